// Proximity_52613349376865
// MI455X (gfx1250) — compile-verified
//
#include <hip/hip_runtime.h>

typedef __attribute__((ext_vector_type(2))) float v2f;
typedef __attribute__((ext_vector_type(8))) float v8f;

#define CLAMP_MIN 1e-12f
#define CLAMP_MAX 1e12f

// One wave == one 16-row tile. The K-reduction (sum over D of (x-c)^2) runs on
// the matrix pipe: V_WMMA_F32_16X16X4_F32 with an all-ones B matrix is a
// 16-row x 4-col row-sum accumulator. A-matrix f32 16x4 layout (ISA 7.12.2):
//   a0: lanes 0-15 = A[m][0], lanes 16-31 = A[m][2]
//   a1: lanes 0-15 = A[m][1], lanes 16-31 = A[m][3]
// Lane L<16 streams cols c..c+3 of row L; lane L+16 streams cols c+4..c+7 of
// the same row. Two WMMAs consume one float4 per lane; since B==1 and we only
// accumulate, K-slot placement is order-independent -> no permutes needed and
// every global load stays 128-bit.
__global__ __launch_bounds__(256) void proximity_tile_kernel(
    const float* __restrict__ x, const int* __restrict__ labels,
    const float* __restrict__ centers, float* __restrict__ partials,
    int numTiles, int D)
{
  const int lane = threadIdx.x & 31;
  const int wave = threadIdx.x >> 5;
  const int tile = blockIdx.x * (blockDim.x >> 5) + wave;
  if (tile >= numTiles) return;

  const int row    = tile * 16 + (lane & 15);
  const int colOff = (lane >> 4) << 2;          // lanes 16-31 start 4 cols later
  const long lab   = (long)labels[row];
  const float* __restrict__ xr = x + (long)row * (long)D + colOff;
  const float* __restrict__ cr = centers + lab * (long)D + colOff;

  v8f acc = {};                                 // C/D: rowsum[m] replicated per column
  const v2f ones = {1.0f, 1.0f};                // B matrix: 4x16 of 1.0f

  for (int c = 0; c < D; c += 8) {
    // Each row's 128B cacheline spans 4 iterations (32B/iter) -> prefetch the
    // x stream once per line, ~1KB ahead, with max locality so it lands in
    // the near caches too (centers are L2-resident already, no prefetch).
    if ((c & 31) == 0)
      __builtin_prefetch(xr + c + 256, 0, 3);
    const float4 xv = *(const float4*)(xr + c);
    const float4 cv = *(const float4*)(cr + c);
    const float dx = xv.x - cv.x;
    const float dy = xv.y - cv.y;
    const float dz = xv.z - cv.z;
    const float dw = xv.w - cv.w;
    v2f a0 = {dx * dx, dy * dy};                // cols {c,c+1 | c+4,c+5}
    v2f a1 = {dz * dz, dw * dw};                // cols {c+2,c+3 | c+6,c+7}
    acc = __builtin_amdgcn_wmma_f32_16x16x4_f32(false, a0, false, ones,
                                                (short)0, acc, false, false);
    acc = __builtin_amdgcn_wmma_f32_16x16x4_f32(false, a1, false, ones,
                                                (short)0, acc, false, false);
  }

  // acc[j] @ lane n(0-15)  = rowsum of tile row j      (M = j)
  // acc[j] @ lane n+16     = rowsum of tile row j+8    (M = j+8)
  // Clamp per row, then fold: 8 adds + one xor-16 shuffle.
  float s = 0.0f;
#pragma unroll
  for (int j = 0; j < 8; ++j) {
    float r = acc[j];
    r = fminf(fmaxf(r, CLAMP_MIN), CLAMP_MAX);
    s += r;
  }
  s += __shfl_xor(s, 16, 32);                   // low half + high half of the tile
  if (lane == 0) partials[tile] = s;            // deterministic: one writer per slot
}

// Deterministic fixed-order tree reduction of the per-tile partials -> mean.
__global__ __launch_bounds__(256) void proximity_finalize_kernel(
    const float* __restrict__ partials, int n, float invB, float* __restrict__ out)
{
  __shared__ float sm[256];
  float s = 0.0f;
  for (int i = threadIdx.x; i < n; i += 256) s += partials[i];
  sm[threadIdx.x] = s;
  __syncthreads();
#pragma unroll
  for (int st = 128; st > 0; st >>= 1) {
    if ((int)threadIdx.x < st) sm[threadIdx.x] += sm[threadIdx.x + st];
    __syncthreads();
  }
  if (threadIdx.x == 0) out[0] = sm[0] * invB;
}

extern "C" void kernel_launch(void* const* d_in, const int* in_sizes, int n_in,
                              void* d_out, int out_size, void* d_ws, size_t ws_size,
                              hipStream_t stream) {
  const float* x       = (const float*)d_in[0];   // [B, D] f32
  const int*   labels  = (const int*)d_in[1];     // [B] int
  const float* centers = (const float*)d_in[2];   // [C, D] f32
  float* out = (float*)d_out;

  const int B = in_sizes[1];                      // 65536
  const int D = in_sizes[0] / B;                  // 512
  const int numTiles = B / 16;                    // 4096 (B % 16 == 0)
  float* partials = (float*)d_ws;                 // 16 KB scratch

  const int wavesPerBlock = 8;                    // 256 threads = 8 wave32
  dim3 block(32 * wavesPerBlock);
  dim3 grid((numTiles + wavesPerBlock - 1) / wavesPerBlock);
  proximity_tile_kernel<<<grid, block, 0, stream>>>(x, labels, centers,
                                                    partials, numTiles, D);
  proximity_finalize_kernel<<<1, 256, 0, stream>>>(partials, numTiles,
                                                   1.0f / (float)B, out);
}